// FeedbackModelNeo_74311524155794
// MI455X (gfx1250) — compile-verified
//
#include <hip/hip_runtime.h>
#include <hip/hip_bf16.h>
#include <math.h>

#define BSZ 16
#define SEQ 2048
#define HID 1024
#define ND  32     // spans per batch
#define NC  8      // discourse types
#define NL  3      // labels

typedef __attribute__((ext_vector_type(2))) float v2f;
typedef __attribute__((ext_vector_type(8))) float v8f;

// ---------------------------------------------------------------------------
// Kernel 1: masked mean pooling over spans. One block per (b,d) span.
// Streams 128 MB of encoder_layer with coalesced float4 loads (HBM-bound:
// ~5.5 us at 23.3 TB/s — this kernel sets total runtime).
// ---------------------------------------------------------------------------
__global__ void __launch_bounds__(256)
pool_kernel(const float* __restrict__ enc,
            const int* __restrict__ head,
            const int* __restrict__ tail,
            float* __restrict__ pooled) {
    const int blk = blockIdx.x;          // b*ND + d
    const int b   = blk / ND;
    const int tid = threadIdx.x;         // 0..255 -> float4 column
    const int s0  = head[blk] + 1;
    const int s1  = tail[blk];

    const float4* base = (const float4*)(enc) + (size_t)b * SEQ * (HID / 4);
    float4 acc = make_float4(0.f, 0.f, 0.f, 0.f);
    for (int s = s0; s < s1; ++s) {
        float4 v = base[(size_t)s * (HID / 4) + tid];
        acc.x += v.x; acc.y += v.y; acc.z += v.z; acc.w += v.w;
    }
    const float inv = 1.0f / (float)(s1 - s0);
    acc.x *= inv; acc.y *= inv; acc.z *= inv; acc.w *= inv;
    ((float4*)pooled)[(size_t)blk * (HID / 4) + tid] = acc;
}

// ---------------------------------------------------------------------------
// Kernel 2: per-batch GEMM pooled(32x1024) x Wr(1024x24->32) via
// V_WMMA_F32_16X16X4_F32. 4 waves per block split the K dimension (256 each);
// per-wave K offsets are baked into the base pointers so the loop has a
// compile-time-constant trip count (64) and stays scalar-controlled (no EXEC
// masking in the hot loop). Partial tiles are combined with LDS float atomics
// (ds_add_f32), then the first 32 lanes gather the selected discourse type,
// do log-softmax over L=3, and accumulate the NLL loss.
//
// WMMA f32 16x16x4 VGPR layouts (ISA 7.12.2):
//   A 16x4 : lane L holds row M=L&15, K = (L>>4)*2 + {0,1} in {v0,v1}
//   B 4x16 : lane L holds col N=L&15, K = (L>>4)*2 + {0,1} in {v0,v1}
//   C/D    : VGPR v holds M = v + (L>>4)*8, N = L&15
//
// Columns 24..31 of the N=1 tile are never read by the gather, so lanes whose
// column is out of range load column 0 instead (no divergence; junk lands in
// dead tile columns).
// ---------------------------------------------------------------------------
__global__ void __launch_bounds__(128)
logits_kernel(const float* __restrict__ pooled,
              const float* __restrict__ Wmat,   // (C,H,L) = (8,1024,3)
              const float* __restrict__ bias,   // (C,L)
              const int* __restrict__ dtid,     // (BS,ND)
              const int* __restrict__ labels,   // (BS,ND)
              float* __restrict__ out,          // 1536 logits + 1 loss slot
              float* __restrict__ lacc) {       // [0]=sum nll, [1]=sum valid
    __shared__ float tile[ND][33];              // 32x32 logits, padded

    const int b    = blockIdx.x;
    const int t    = threadIdx.x;               // 0..127
    const int wave = t >> 5;                    // 0..3: K-chunk owner
    const int lane = t & 31;
    const int lo   = lane & 15;
    const int hi   = lane >> 4;                 // 0 or 1

    // Zero the shared accumulation tile.
    for (int i = t; i < ND * 33; i += 128) ((float*)tile)[i] = 0.f;
    __syncthreads();

    // Per-wave K chunk: 256 K-values, baked into base pointers.
    const int kbeg = wave * (HID / 4);
    const int kofs = kbeg + hi * 2;             // + lane-half K offset

    // A rows for the two M tiles.
    const float* pA0 = pooled + ((size_t)b * ND + lo) * HID + kofs;
    const float* pA1 = pA0 + (size_t)16 * HID;

    // B columns for the two N tiles: col -> (c = col/3, l = col%3),
    // B[k][col] = W[c, k, l] = Wmat[(c*HID + k)*NL + l]. Clamp dead columns
    // to column 0 so all loads are unconditional.
    const int col0 = lo;                                    // 0..15, valid
    const int col1 = (16 + lo < NC * NL) ? (16 + lo) : 0;   // 16..23 valid, else 0
    const float* pB0 = Wmat + (size_t)(col0 / NL) * HID * NL + (col0 % NL)
                            + (size_t)kofs * NL;
    const float* pB1 = Wmat + (size_t)(col1 / NL) * HID * NL + (col1 % NL)
                            + (size_t)kofs * NL;

    v8f a00 = {0,0,0,0,0,0,0,0};
    v8f a01 = {0,0,0,0,0,0,0,0};
    v8f a10 = {0,0,0,0,0,0,0,0};
    v8f a11 = {0,0,0,0,0,0,0,0};

    #pragma unroll 4
    for (int k0 = 0; k0 < HID / 4; k0 += 4) {   // constant trip count: 64
        v2f avec0 = *(const v2f*)(pA0 + k0);
        v2f avec1 = *(const v2f*)(pA1 + k0);
        v2f bvec0, bvec1;
        bvec0.x = pB0[(size_t)k0 * NL];
        bvec0.y = pB0[(size_t)(k0 + 1) * NL];
        bvec1.x = pB1[(size_t)k0 * NL];
        bvec1.y = pB1[(size_t)(k0 + 1) * NL];

        a00 = __builtin_amdgcn_wmma_f32_16x16x4_f32(false, avec0, false, bvec0,
                                                    (short)0, a00, false, false);
        a01 = __builtin_amdgcn_wmma_f32_16x16x4_f32(false, avec0, false, bvec1,
                                                    (short)0, a01, false, false);
        a10 = __builtin_amdgcn_wmma_f32_16x16x4_f32(false, avec1, false, bvec0,
                                                    (short)0, a10, false, false);
        a11 = __builtin_amdgcn_wmma_f32_16x16x4_f32(false, avec1, false, bvec1,
                                                    (short)0, a11, false, false);
    }

    // Combine the 4 waves' partial tiles via LDS float atomics.
    // Element v of lane -> row (mtile*16 + hi*8 + v), col (ntile*16 + lo).
    #pragma unroll
    for (int v = 0; v < 8; ++v) {
        atomicAdd(&tile[hi * 8 + v][lo],           a00[v]);
        atomicAdd(&tile[hi * 8 + v][16 + lo],      a01[v]);
        atomicAdd(&tile[16 + hi * 8 + v][lo],      a10[v]);
        atomicAdd(&tile[16 + hi * 8 + v][16 + lo], a11[v]);
    }
    __syncthreads();

    // First 32 lanes: one span d = t each. Gather, softmax, NLL.
    if (t < ND) {
        const int gd  = b * ND + t;
        const int c   = dtid[gd];
        const int lab = labels[gd];
        const float v0 = tile[t][c * NL + 0] + bias[c * NL + 0];
        const float v1 = tile[t][c * NL + 1] + bias[c * NL + 1];
        const float v2 = tile[t][c * NL + 2] + bias[c * NL + 2];
        out[gd * NL + 0] = v0;
        out[gd * NL + 1] = v1;
        out[gd * NL + 2] = v2;

        const float mx  = fmaxf(v0, fmaxf(v1, v2));
        const float lse = mx + __logf(__expf(v0 - mx) + __expf(v1 - mx) + __expf(v2 - mx));
        const int   sl  = (lab >= 0) ? lab : 0;
        const float sel = (sl == 0) ? v0 : ((sl == 1) ? v1 : v2);
        if (lab >= 0) {
            atomicAdd(&lacc[0], lse - sel);
            atomicAdd(&lacc[1], 1.0f);
        }
    }
}

// ---------------------------------------------------------------------------
// Kernel 3: loss = sum(nll) / sum(valid)
// ---------------------------------------------------------------------------
__global__ void finalize_kernel(const float* __restrict__ lacc,
                                float* __restrict__ out) {
    out[BSZ * ND * NL] = lacc[0] / lacc[1];
}

extern "C" void kernel_launch(void* const* d_in, const int* in_sizes, int n_in,
                              void* d_out, int out_size, void* d_ws, size_t ws_size,
                              hipStream_t stream) {
    const float* enc   = (const float*)d_in[0];   // (BS,S,H) f32
    const float* Wmat  = (const float*)d_in[1];   // (C,H,L)  f32
    const float* bias  = (const float*)d_in[2];   // (C,L)    f32
    const int*   head  = (const int*)d_in[3];     // (BS,D)
    const int*   tail  = (const int*)d_in[4];     // (BS,D)
    const int*   dtid  = (const int*)d_in[5];     // (BS,D)
    const int*   labl  = (const int*)d_in[6];     // (BS,D)
    float* out = (float*)d_out;                   // 1536 logits + 1 loss

    float* lacc   = (float*)d_ws;                 // 2 floats (loss accum)
    float* pooled = (float*)((char*)d_ws + 256);  // BS*ND*HID f32 = 2 MB

    hipMemsetAsync(d_ws, 0, 256, stream);
    pool_kernel<<<BSZ * ND, 256, 0, stream>>>(enc, head, tail, pooled);
    logits_kernel<<<BSZ, 128, 0, stream>>>(pooled, Wmat, bias, dtid, labl, out, lacc);
    finalize_kernel<<<1, 1, 0, stream>>>(lacc, out);
}